// SimpleContrastiveLoss_65008624992463
// MI455X (gfx1250) — compile-verified
//
#include <hip/hip_runtime.h>

typedef float v2f __attribute__((ext_vector_type(2)));
typedef float v8f __attribute__((ext_vector_type(8)));

#define N_ROWS 4096
#define DIM    512
#define MARGIN_F 0.2f

#define TM 128          // rows of sim per workgroup
#define TN 64           // cols of sim per workgroup
#define KB 16           // K-step staged through LDS
#define LDST 20         // LDS row stride in floats (16B-aligned rows, conflict-free)

// ---------------------------------------------------------------------------
// Kernel 1: pos[i] = <image_i, text_i>; also zero the fp64 accumulator.
// ---------------------------------------------------------------------------
__global__ __launch_bounds__(256) void pos_init_kernel(const float* __restrict__ img,
                                                       const float* __restrict__ txt,
                                                       float* __restrict__ pos,
                                                       double* __restrict__ accum) {
  int i = blockIdx.x * blockDim.x + threadIdx.x;
  if (i == 0) *accum = 0.0;
  if (i < N_ROWS) {
    const float4* a = (const float4*)(img + (size_t)i * DIM);
    const float4* b = (const float4*)(txt + (size_t)i * DIM);
    float s = 0.f;
#pragma unroll 4
    for (int k = 0; k < DIM / 4; ++k) {
      float4 x = a[k];
      float4 y = b[k];
      s += x.x * y.x + x.y * y.y + x.z * y.z + x.w * y.w;
    }
    pos[i] = s;
  }
}

// ---------------------------------------------------------------------------
// Kernel 2: WMMA-tiled GEMM (sim tile) fused with margin-ReLU epilogue and
// hierarchical reduction to a single fp64 accumulator.
//   grid = (N/TM, N/TN), block = 256 threads = 8 waves (wave32).
//   Wave w owns rows [w*16, w*16+16) of the 128-row tile, all 64 cols.
// ---------------------------------------------------------------------------
__global__ __launch_bounds__(256) void gemm_loss_kernel(const float* __restrict__ img,
                                                        const float* __restrict__ txt,
                                                        const float* __restrict__ pos,
                                                        double* __restrict__ accum) {
  __shared__ float lA[TM * LDST];   // 128 x 16 tile of image (padded rows)
  __shared__ float lB[TN * LDST];   // 64  x 16 tile of text  (padded rows)
  __shared__ float red[8];

  const int tid  = threadIdx.x;
  const int wid  = tid >> 5;        // wave id 0..7
  const int lane = tid & 31;
  const int lr   = lane & 15;       // lane-in-half
  const int half = lane >> 4;       // 0 / 1

  const int rowBase = blockIdx.x * TM;
  const int colBase = blockIdx.y * TN;

  v8f acc[4] = {};                  // 4 column subtiles of 16x16 f32

  for (int k0 = 0; k0 < DIM; k0 += KB) {
    __syncthreads();                // protect LDS from previous iteration's reads

    // --- stage A tile: 128x16 floats, 2 float4 per thread ---
#pragma unroll
    for (int i = 0; i < 2; ++i) {
      int fid = tid * 2 + i;        // 0..511 float4 ids
      int row = fid >> 2;           // 4 float4 per row
      int kc  = (fid & 3) << 2;
      float4 v = *(const float4*)(img + (size_t)(rowBase + row) * DIM + k0 + kc);
      *(float4*)&lA[row * LDST + kc] = v;
    }
    // --- stage B tile: 64x16 floats, 1 float4 per thread ---
    {
      int row = tid >> 2;
      int kc  = (tid & 3) << 2;
      float4 v = *(const float4*)(txt + (size_t)(colBase + row) * DIM + k0 + kc);
      *(float4*)&lB[row * LDST + kc] = v;
    }
    __syncthreads();

    // --- 4 K-substeps of 4, 4 column subtiles each -> 16 WMMAs ---
    // fp32 A 16x4 layout: lane = half*16 + M, VGPR v holds K = 2*half + v
    // fp32 B 4x16 layout: lane = half*16 + N, VGPR v holds K = 2*half + v
#pragma unroll
    for (int t = 0; t < 4; ++t) {
      const int kk = 4 * t + 2 * half;
      v2f a  = *(const v2f*)&lA[(wid * 16 + lr) * LDST + kk];
      v2f b0 = *(const v2f*)&lB[( 0 + lr) * LDST + kk];
      v2f b1 = *(const v2f*)&lB[(16 + lr) * LDST + kk];
      v2f b2 = *(const v2f*)&lB[(32 + lr) * LDST + kk];
      v2f b3 = *(const v2f*)&lB[(48 + lr) * LDST + kk];
      acc[0] = __builtin_amdgcn_wmma_f32_16x16x4_f32(false, a, false, b0, (short)0, acc[0], false, false);
      acc[1] = __builtin_amdgcn_wmma_f32_16x16x4_f32(false, a, false, b1, (short)0, acc[1], false, false);
      acc[2] = __builtin_amdgcn_wmma_f32_16x16x4_f32(false, a, false, b2, (short)0, acc[2], false, false);
      acc[3] = __builtin_amdgcn_wmma_f32_16x16x4_f32(false, a, false, b3, (short)0, acc[3], false, false);
    }
  }

  // --- epilogue: relu(MARGIN - pos[i] + sim[i,j]), mask diagonal ---
  // C/D layout: lanes 0-15: M = v,   N = lane
  //             lanes16-31: M = 8+v, N = lane-16
  float pcache[8];
#pragma unroll
  for (int v = 0; v < 8; ++v)
    pcache[v] = pos[rowBase + wid * 16 + half * 8 + v];

  float partial = 0.f;
#pragma unroll
  for (int c = 0; c < 4; ++c) {
    const int gj = colBase + c * 16 + lr;
#pragma unroll
    for (int v = 0; v < 8; ++v) {
      const int gi = rowBase + wid * 16 + half * 8 + v;
      float m = MARGIN_F - pcache[v] + acc[c][v];
      m = fmaxf(m, 0.f);
      if (gi == gj) m = 0.f;
      partial += m;
    }
  }

  // wave32 reduction
#pragma unroll
  for (int off = 16; off > 0; off >>= 1)
    partial += __shfl_xor(partial, off, 32);
  if (lane == 0) red[wid] = partial;
  __syncthreads();
  if (tid == 0) {
    double s = 0.0;
#pragma unroll
    for (int w = 0; w < 8; ++w) s += (double)red[w];
    atomicAdd(accum, s);   // global_atomic_add_f64
  }
}

// ---------------------------------------------------------------------------
// Kernel 3: scale and emit the fp32 scalar.
// ---------------------------------------------------------------------------
__global__ void finalize_kernel(const double* __restrict__ accum, float* __restrict__ out) {
  out[0] = (float)(*accum / ((double)N_ROWS * (double)(N_ROWS - 1)));
}

extern "C" void kernel_launch(void* const* d_in, const int* in_sizes, int n_in,
                              void* d_out, int out_size, void* d_ws, size_t ws_size,
                              hipStream_t stream) {
  (void)in_sizes; (void)n_in; (void)out_size; (void)ws_size;
  const float* img = (const float*)d_in[0];
  const float* txt = (const float*)d_in[1];
  float* out = (float*)d_out;

  double* accum = (double*)d_ws;                     // 8B (16B-aligned slot)
  float*  pos   = (float*)((char*)d_ws + 16);        // 4096 floats

  pos_init_kernel<<<N_ROWS / 256, 256, 0, stream>>>(img, txt, pos, accum);
  gemm_loss_kernel<<<dim3(N_ROWS / TM, N_ROWS / TN), 256, 0, stream>>>(img, txt, pos, accum);
  finalize_kernel<<<1, 1, 0, stream>>>(accum, out);
}